// VocoMorph_21328807591983
// MI455X (gfx1250) — compile-verified
//
#include <hip/hip_runtime.h>
#include <math.h>

// ---------------------------------------------------------------------------
// VocoMorph on MI455X (gfx1250): STFT/iSTFT expressed as f32 WMMA matmuls.
//   spec  = frames(13x1024) x [Fr|Fi](1024x513)      (window folded into F)
//   y     = Xr(13x513) x Gr + Xi(13x513) x Gi        (window + 1/N + Hermitian
//                                                     symmetry folded into G)
// FiLM modulation is a real per-bin gain: mod_spec = spec*(gamma + beta/mag).
//
// v2: 2 chunks per workgroup -> each DFT-operator load feeds 4 WMMA chains.
// v3: 6-argument tensor_load_to_lds builtin (this toolchain's arity).
// v4: WGP-scope prefetch (locality 3) + software double-buffered inner loops
//     so operand loads for step i+1 issue before the WMMA group of step i.
// ---------------------------------------------------------------------------

typedef float v2f __attribute__((ext_vector_type(2)));
typedef float v8f __attribute__((ext_vector_type(8)));
typedef unsigned int v4u __attribute__((ext_vector_type(4)));
typedef int v4i __attribute__((ext_vector_type(4)));
typedef int v8i __attribute__((ext_vector_type(8)));

#define CHUNKN   4096
#define STRIDEN  3072
#define NFFT     1024
#define HOP      256
#define NFREQ    513
#define NFRAMES  13
#define NBATCH   8
#define TLEN     (512 * STRIDEN)
#define NCHUNKS  (TLEN / STRIDEN)
#define CPB      2                         // chunks per workgroup

#define KPAD     528                       // NFREQ padded to multiple of 16
#define FPAD     16                        // NFRAMES padded to 16
#define CHUNKPAD (HOP * (FPAD - 1) + NFFT) // 4864 (A rows 13..15 read this tail)

// workspace layout (float offsets); total ~8.7 MB
#define OFF_GB   0
#define SZ_GB    8448                      // 8 x 1026 (gamma||beta), rounded
#define SZ_OP    (KPAD * NFFT)             // 540672 (same extent for F and G)
#define OFF_FFR  (OFF_GB + SZ_GB)
#define OFF_FFI  (OFF_FFR + SZ_OP)
#define OFF_GTR  (OFF_FFI + SZ_OP)
#define OFF_GTI  (OFF_GTR + SZ_OP)
#define OFF_ENV  (OFF_GTI + SZ_OP)

#if defined(__HIP_DEVICE_COMPILE__) && defined(__has_builtin)
#if __has_builtin(__builtin_amdgcn_tensor_load_to_lds) && \
    __has_builtin(__builtin_amdgcn_s_wait_tensorcnt)
#define USE_TDM 1
#endif
#endif

__device__ __forceinline__ v8f wmma_f32(v2f a, v2f b, v8f c) {
  // D = A(16x4,f32) * B(4x16,f32) + C(16x16,f32)
  return __builtin_amdgcn_wmma_f32_16x16x4_f32(false, a, false, b,
                                               (short)0, c, false, false);
}

#ifdef USE_TDM
// 1-D TDM copy: nelem_tile f32 elements global -> LDS.  tensor_dim0 is the
// number of in-bounds elements; TDM returns zeros past the tensor edge, which
// implements the reference's zero padding of the final chunk for free.
__device__ __forceinline__ void tdm_load_1d(const float* gsrc, float* ldst,
                                            unsigned nelem_tile,
                                            unsigned nelem_tensor) {
  const unsigned long long ga = (unsigned long long)(uintptr_t)gsrc;
  const unsigned lds = (unsigned)(uintptr_t)ldst;  // low 32b = LDS offset
  v4u g0;
  g0[0] = 1u;                                     // count=1, user descriptor
  g0[1] = lds;                                    // lds_addr
  g0[2] = (unsigned)(ga & 0xffffffffu);           // global_addr[31:0]
  g0[3] = (unsigned)((ga >> 32) & 0x01ffffffu) | (2u << 30);  // [56:32]|type=2
  v8i g1;
  g1[0] = (int)(2u << 16);                        // data_size=2 (4B); mask=0
  g1[1] = (int)((nelem_tensor & 0xffffu) << 16);  // tensor_dim0[15:0]
  g1[2] = (int)(((nelem_tensor >> 16) & 0xffffu)  // tensor_dim0[31:16]
                | (1u << 16));                    // tensor_dim1 = 1
  g1[3] = (int)((nelem_tile & 0xffffu) << 16);    // tile_dim0
  g1[4] = 1;                                      // tile_dim1=1, tile_dim2=0
  g1[5] = (int)nelem_tile;                        // tensor_dim0_stride lo
  g1[6] = 0;                                      // stride hi / dim1_stride lo
  g1[7] = 0;
  const v4i z4 = {0, 0, 0, 0};                    // groups 2/3 unused (<=2D)
  const v8i z8 = {0, 0, 0, 0, 0, 0, 0, 0};        // unused trailing group
  __builtin_amdgcn_tensor_load_to_lds(g0, g1, z4, z4, z8, 0);
}
#endif

// ---------------- operator tables: windowed DFT / windowed iDFT ------------
__global__ void k_tables(float* __restrict__ ffr, float* __restrict__ ffi,
                         float* __restrict__ gtr, float* __restrict__ gti,
                         float* __restrict__ env) {
  const int i = blockIdx.x * blockDim.x + threadIdx.x;
  const float TWO_PI = 6.283185307179586f;
  if (i < SZ_OP) {
    // forward operator, stored transposed: ffr[kf][n], stride NFFT
    {
      const int kf = i / NFFT, n = i % NFFT;
      float fr = 0.f, fi = 0.f;
      if (kf < NFREQ) {
        const float wn = 0.5f * (1.0f - cosf(TWO_PI * (float)n / (float)NFFT));
        const int ph = (n * kf) & (NFFT - 1);
        float s, c;
        sincosf(TWO_PI * (float)ph / (float)NFFT, &s, &c);
        fr = wn * c;          // Re{e^{-i th}}
        fi = -wn * s;         // Im{e^{-i th}}
      }
      ffr[i] = fr; ffi[i] = fi;
    }
    // inverse operator, stored transposed: gtr[n][k], stride KPAD
    {
      const int n2 = i / KPAD, k2 = i % KPAD;
      float gr = 0.f, gi = 0.f;
      if (k2 < NFREQ) {
        const float wn = 0.5f * (1.0f - cosf(TWO_PI * (float)n2 / (float)NFFT));
        const float coef =
            ((k2 == 0) || (k2 == NFFT / 2)) ? (1.0f / NFFT) : (2.0f / NFFT);
        const int ph = (n2 * k2) & (NFFT - 1);
        float s, c;
        sincosf(TWO_PI * (float)ph / (float)NFFT, &s, &c);
        gr = coef * wn * c;   // y += Xr*cos
        gi = -coef * wn * s;  // y -= Xi*sin
      }
      gtr[i] = gr; gti[i] = gi;
    }
  }
  if (i < CHUNKN) {           // OLA window-energy envelope (chunk-invariant)
    float e = 0.f;
    for (int f = 0; f < NFRAMES; ++f) {
      const int j = i - f * HOP;
      if (j >= 0 && j < NFFT) {
        const float wn = 0.5f * (1.0f - cosf(TWO_PI * (float)j / (float)NFFT));
        e += wn * wn;
      }
    }
    env[i] = e;
  }
}

// ---------------- FiLM: gb[b] = emb_table[effect_id[b]] @ film_w + film_b --
__global__ void k_film(const int* __restrict__ effect_id,
                       const float* __restrict__ emb,
                       const float* __restrict__ film_w,
                       const float* __restrict__ film_b,
                       float* __restrict__ gb) {
  const int i = blockIdx.x * blockDim.x + threadIdx.x;
  if (i >= NBATCH * 2 * NFREQ) return;
  const int b = i / (2 * NFREQ), j = i % (2 * NFREQ);
  const float* er = emb + effect_id[b] * 128;
  float s = film_b[j];
  for (int t = 0; t < 128; ++t) s += er[t] * film_w[t * (2 * NFREQ) + j];
  gb[b * (2 * NFREQ) + j] = s;
}

__global__ void k_zero(float* __restrict__ p, int n) {
  const int i = blockIdx.x * blockDim.x + threadIdx.x;
  if (i < n) p[i] = 0.f;
}

// ---------------- main: one workgroup per (2 chunks, batch) ----------------
__global__ __launch_bounds__(256) void k_vocomorph(
    const float* __restrict__ audio, const float* __restrict__ gb,
    const float* __restrict__ ffr, const float* __restrict__ ffi,
    const float* __restrict__ gtr, const float* __restrict__ gti,
    const float* __restrict__ env, float* __restrict__ out) {
  extern __shared__ float smem[];
  float* chunkS0 = smem;                      // [CHUNKPAD]
  float* chunkS1 = smem + CHUNKPAD;           // [CHUNKPAD]
  float* acc0    = chunkS0;                   // [CHUNKN] alias (phase 4-5)
  float* acc1    = chunkS1;                   // [CHUNKN] alias (phase 4-5)
  float* sR0     = smem + 2 * CHUNKPAD;       // [FPAD*KPAD]
  float* sI0     = sR0 + FPAD * KPAD;
  float* sR1     = sI0 + FPAD * KPAD;
  float* sI1     = sR1 + FPAD * KPAD;

  const int chunk0 = blockIdx.x * CPB;
  const int b      = blockIdx.y;
  const int tid    = threadIdx.x;
  const int wave   = tid >> 5;                // wave32
  const int lane   = tid & 31;
  const int half   = lane >> 4;               // K-half selector for WMMA A/B
  const int l      = lane & 15;               // M (A) / N (B) index in tile
  const long tbase = (long)chunk0 * STRIDEN;

  // phase 1: stage both chunks into LDS
#ifdef USE_TDM
  if (wave == 0) {                            // TDM is a wave-level DMA op
    tdm_load_1d(audio + (size_t)b * TLEN + tbase, chunkS0, CHUNKPAD,
                (unsigned)(TLEN - tbase));
    tdm_load_1d(audio + (size_t)b * TLEN + tbase + STRIDEN, chunkS1, CHUNKPAD,
                (unsigned)(TLEN - (tbase + STRIDEN)));
    __builtin_amdgcn_s_wait_tensorcnt(0);
  }
#else
  for (int j = tid; j < CHUNKPAD; j += 256) {
    long t0 = tbase + j, t1 = tbase + STRIDEN + j;
    chunkS0[j] = (t0 < TLEN) ? audio[(size_t)b * TLEN + t0] : 0.f;
    chunkS1[j] = (t1 < TLEN) ? audio[(size_t)b * TLEN + t1] : 0.f;
  }
#endif
  __syncthreads();

  // phase 2: forward rFFT.  A[m][k] = chunk[HOP*m + k]; one B load pair feeds
  // 4 WMMA chains (chunk0/chunk1 x re/im); operand loads double-buffered.
  {
    const float* a0Row = chunkS0 + HOP * l + 2 * half;
    const float* a1Row = chunkS1 + HOP * l + 2 * half;
    for (int tile = wave; tile < KPAD / 16; tile += 8) {
      const int kf0 = tile * 16;
      const float* brRow = ffr + (size_t)(kf0 + l) * NFFT + 2 * half;
      const float* biRow = ffi + (size_t)(kf0 + l) * NFFT + 2 * half;
      v8f r0 = {}, i0 = {}, r1 = {}, i1 = {};
      v2f a0c = *(const v2f*)(a0Row);
      v2f a1c = *(const v2f*)(a1Row);
      v2f brc = *(const v2f*)(brRow);
      v2f bic = *(const v2f*)(biRow);
      for (int n0 = 0; n0 < NFFT - 4; n0 += 4) {
        const v2f a0n = *(const v2f*)(a0Row + n0 + 4);
        const v2f a1n = *(const v2f*)(a1Row + n0 + 4);
        const v2f brn = *(const v2f*)(brRow + n0 + 4);
        const v2f bin = *(const v2f*)(biRow + n0 + 4);
        __builtin_prefetch(brRow + n0 + 512, 0, 3);   // WGP-scope prefetch
        __builtin_prefetch(biRow + n0 + 512, 0, 3);
        r0 = wmma_f32(a0c, brc, r0);
        i0 = wmma_f32(a0c, bic, i0);
        r1 = wmma_f32(a1c, brc, r1);
        i1 = wmma_f32(a1c, bic, i1);
        a0c = a0n; a1c = a1n; brc = brn; bic = bin;
      }
      r0 = wmma_f32(a0c, brc, r0);                    // epilogue step
      i0 = wmma_f32(a0c, bic, i0);
      r1 = wmma_f32(a1c, brc, r1);
      i1 = wmma_f32(a1c, bic, i1);
#pragma unroll
      for (int j = 0; j < 8; ++j) {          // D tile: (M=j+8*half, N=kf0+l)
        const int o = (j + 8 * half) * KPAD + kf0 + l;
        sR0[o] = r0[j]; sI0[o] = i0[j];
        sR1[o] = r1[j]; sI1[o] = i1[j];
      }
    }
  }
  __syncthreads();

  // phase 3: FiLM modulation (real per-bin gain); rows f>=13 and bins k>=513
  // are forced to exactly 0 (this also cleans any junk the chunk-tail rows
  // produced).  Then zero the OLA accumulators (chunk buffers are dead).
  for (int cc = 0; cc < CPB; ++cc) {
    float* sR = cc ? sR1 : sR0;
    float* sI = cc ? sI1 : sI0;
    for (int e = tid; e < FPAD * KPAD; e += 256) {
      const int f = e / KPAD, k = e - f * KPAD;
      const float xr = sR[e], xi = sI[e];
      float orr = 0.f, oii = 0.f;
      if (f < NFRAMES && k < NFREQ) {
        const float gamma = gb[b * (2 * NFREQ) + k];
        const float beta  = gb[b * (2 * NFREQ) + NFREQ + k];
        const float mag = sqrtf(xr * xr + xi * xi);
        if (mag > 0.f) {
          const float s = gamma + beta / mag;
          orr = xr * s; oii = xi * s;
        } else {
          orr = beta;                        // phase==0 branch of reference
        }
      }
      sR[e] = orr; sI[e] = oii;
    }
  }
  for (int j = tid; j < CHUNKN; j += 256) { acc0[j] = 0.f; acc1[j] = 0.f; }
  __syncthreads();

  // phase 4: inverse rFFT + intra-chunk overlap-add (LDS float atomics).
  // One G-operator load pair feeds 4 WMMA chains; double-buffered likewise.
  {
    const float* ar0 = sR0 + KPAD * l + 2 * half;
    const float* ai0 = sI0 + KPAD * l + 2 * half;
    const float* ar1 = sR1 + KPAD * l + 2 * half;
    const float* ai1 = sI1 + KPAD * l + 2 * half;
    for (int tile = wave; tile < NFFT / 16; tile += 8) {
      const int n0c = tile * 16;
      const float* brRow = gtr + (size_t)(n0c + l) * KPAD + 2 * half;
      const float* biRow = gti + (size_t)(n0c + l) * KPAD + 2 * half;
      v8f d0 = {}, d1 = {};
      v2f xr0c = *(const v2f*)(ar0);
      v2f xi0c = *(const v2f*)(ai0);
      v2f xr1c = *(const v2f*)(ar1);
      v2f xi1c = *(const v2f*)(ai1);
      v2f brc  = *(const v2f*)(brRow);
      v2f bic  = *(const v2f*)(biRow);
      for (int k0 = 0; k0 < KPAD - 4; k0 += 4) {
        const v2f xr0n = *(const v2f*)(ar0 + k0 + 4);
        const v2f xi0n = *(const v2f*)(ai0 + k0 + 4);
        const v2f xr1n = *(const v2f*)(ar1 + k0 + 4);
        const v2f xi1n = *(const v2f*)(ai1 + k0 + 4);
        const v2f brn  = *(const v2f*)(brRow + k0 + 4);
        const v2f bin  = *(const v2f*)(biRow + k0 + 4);
        __builtin_prefetch(brRow + k0 + 512, 0, 3);
        __builtin_prefetch(biRow + k0 + 512, 0, 3);
        d0 = wmma_f32(xr0c, brc, d0);
        d0 = wmma_f32(xi0c, bic, d0);
        d1 = wmma_f32(xr1c, brc, d1);
        d1 = wmma_f32(xi1c, bic, d1);
        xr0c = xr0n; xi0c = xi0n; xr1c = xr1n; xi1c = xi1n;
        brc = brn; bic = bin;
      }
      d0 = wmma_f32(xr0c, brc, d0);                   // epilogue step
      d0 = wmma_f32(xi0c, bic, d0);
      d1 = wmma_f32(xr1c, brc, d1);
      d1 = wmma_f32(xi1c, bic, d1);
#pragma unroll
      for (int j = 0; j < 8; ++j) {
        const int m = j + 8 * half;          // frame index of this D row
        if (m < NFRAMES) {
          atomicAdd(&acc0[m * HOP + n0c + l], d0[j]);
          atomicAdd(&acc1[m * HOP + n0c + l], d1[j]);
        }
      }
    }
  }
  __syncthreads();

  // phase 5: env normalization + inter-chunk overlap-add into global output
  for (int cc = 0; cc < CPB; ++cc) {
    const float* acc = cc ? acc1 : acc0;
    const long tb = tbase + (long)cc * STRIDEN;
    for (int j = tid; j < CHUNKN; j += 256) {
      const long t = tb + j;
      if (t < TLEN) {
        const float v = acc[j] / fmaxf(env[j], 1e-11f);
        atomicAdd(&out[(size_t)b * TLEN + t], v);
      }
    }
  }
}

// ---------------------------------------------------------------------------
extern "C" void kernel_launch(void* const* d_in, const int* in_sizes, int n_in,
                              void* d_out, int out_size, void* d_ws,
                              size_t ws_size, hipStream_t stream) {
  const int*   effect_id = (const int*)d_in[0];
  const float* audio     = (const float*)d_in[1];
  const float* emb       = (const float*)d_in[2];
  const float* film_w    = (const float*)d_in[3];
  const float* film_b    = (const float*)d_in[4];
  // d_in[5] = window; recomputed analytically (exact Hann formula)
  float* out = (float*)d_out;
  float* ws  = (float*)d_ws;   // requires ~8.7 MB of scratch

  float* gb  = ws + OFF_GB;
  float* ffr = ws + OFF_FFR;
  float* ffi = ws + OFF_FFI;
  float* gtr = ws + OFF_GTR;
  float* gti = ws + OFF_GTI;
  float* env = ws + OFF_ENV;

  k_tables<<<(SZ_OP + 255) / 256, 256, 0, stream>>>(ffr, ffi, gtr, gti, env);
  k_film<<<(NBATCH * 2 * NFREQ + 255) / 256, 256, 0, stream>>>(
      effect_id, emb, film_w, film_b, gb);
  k_zero<<<(out_size + 255) / 256, 256, 0, stream>>>(out, out_size);

  dim3 grid(NCHUNKS / CPB, NBATCH);
  const size_t smem_bytes =
      (size_t)(CPB * CHUNKPAD + 2 * CPB * FPAD * KPAD) * sizeof(float);
  k_vocomorph<<<grid, 256, smem_bytes, stream>>>(audio, gb, ffr, ffi, gtr, gti,
                                                 env, out);
}